// DSSKernel_35055523070058
// MI455X (gfx1250) — compile-verified
//
#include <hip/hip_runtime.h>

// DSS kernel for MI455X (gfx1250, wave32).
//
// out[l] = Re( sum_n c[n] * exp(lam_n * l) ),  lam_n = step*(Lre[n] + i*Lim[n])
// c[n] = (W[n]/Lam[n]) * conj(s_n)/(|s_n|^2 + eps),  s_n = geometric series (closed form).
//
// With l = l0 + 16*m + j, each channel's contribution factors exactly into
// (m,k)-dependent times (k,j)-dependent terms (angle/exponent addition), so a
// 16x16 output tile is two 16x16x256 f32 GEMMs -> V_WMMA_F32_16X16X4_F32.

typedef __attribute__((ext_vector_type(2))) float v2f;
typedef __attribute__((ext_vector_type(8))) float v8f;

#define N_CH 256
#define L_TOT 65536
#define DSS_EPS 1e-7f

// ---------------------------------------------------------------------------
// Kernel 1: per-channel coefficients. ws layout (floats):
//   [0,256)   c_re   [256,512) c_im   [512,768) a = step*Lre   [768,1024) b = step*Lim
// ---------------------------------------------------------------------------
__global__ __launch_bounds__(N_CH) void dss_prep_kernel(
    const float* __restrict__ W, const float* __restrict__ log_step,
    const float* __restrict__ Lre_p, const float* __restrict__ Lim_p,
    float* __restrict__ ws)
{
  int n = threadIdx.x;
  float step = __expf(log_step[0]);
  float Lre = Lre_p[n], Lim = Lim_p[n];
  float a = step * Lre;          // < 0
  float b = step * Lim;

  // r = exp(a + i b)
  float sb, cb; __sincosf(b, &sb, &cb);
  float ea = __expf(a);
  float r_re = ea * cb, r_im = ea * sb;

  // r^L (underflows to ~0; kept for generality)
  float sL, cL; __sincosf(b * (float)L_TOT, &sL, &cL);
  float eL = __expf(a * (float)L_TOT);
  float rL_re = eL * cL, rL_im = eL * sL;

  // s = (1 - r^L) / (1 - r)
  float num_re = 1.0f - rL_re, num_im = -rL_im;
  float den_re = 1.0f - r_re,  den_im = -r_im;
  float dmag = den_re * den_re + den_im * den_im;
  float s_re = (num_re * den_re + num_im * den_im) / dmag;
  float s_im = (num_im * den_re - num_re * den_im) / dmag;

  // recip = conj(s) / (|s|^2 + eps)
  float smag = s_re * s_re + s_im * s_im + DSS_EPS;
  float rc_re = s_re / smag, rc_im = -s_im / smag;

  // coef = Wc / Lam
  float wre = W[2 * n], wim = W[2 * n + 1];
  float lmag = Lre * Lre + Lim * Lim;
  float co_re = (wre * Lre + wim * Lim) / lmag;
  float co_im = (wim * Lre - wre * Lim) / lmag;

  // c = coef * recip
  ws[n]            = co_re * rc_re - co_im * rc_im;
  ws[N_CH + n]     = co_re * rc_im + co_im * rc_re;
  ws[2 * N_CH + n] = a;
  ws[3 * N_CH + n] = b;
}

// ---------------------------------------------------------------------------
// Kernel 2: 8 waves/block, each wave produces one 16x16 tile of out[] via
// 2 GEMMs of shape 16x16x256 built from V_WMMA_F32_16X16X4_F32 (K-step 4).
// ---------------------------------------------------------------------------
__global__ __launch_bounds__(256) void dss_wmma_kernel(
    const float* __restrict__ ws, float* __restrict__ out)
{
  __shared__ float lds[4 * N_CH];
  int tid = threadIdx.x;
  // Stage channel constants into LDS once per block.
  lds[tid]       = ws[tid];
  lds[tid + 256] = ws[tid + 256];
  lds[tid + 512] = ws[tid + 512];
  lds[tid + 768] = ws[tid + 768];
  __syncthreads();

  const float* lcr = lds;
  const float* lci = lds + 256;
  const float* la  = lds + 512;
  const float* lb  = lds + 768;

  int wave  = tid >> 5;        // 0..7
  int lane  = tid & 31;
  int half  = lane >> 4;       // selects K pair within the 16x4 / 4x16 fragment
  int idx16 = lane & 15;       // M (A-side) / N (B-side) index
  int tile  = blockIdx.x * 8 + wave;
  int l0    = tile * 256;

  float pA = (float)(l0 + 16 * idx16);   // l-position carried by the A fragment
  float pB = (float)idx16;               // l-position carried by the B fragment

  v8f acc = {};
  for (int kk = 0; kk < N_CH; kk += 4) {
    v2f a1, a2, b1, b2;
#pragma unroll
    for (int t = 0; t < 2; ++t) {
      int k = kk + 2 * half + t;         // ISA A/B fragment layout: VGPR t holds K = 2*half + t
      float ak = la[k], bk = lb[k];
      float cr = lcr[k], ci = lci[k];

      float sA, cA; __sincosf(bk * pA, &sA, &cA);
      float eA = __expf(ak * pA);
      float A1 = eA * (cr * cA - ci * sA);
      float A2 = -eA * (cr * sA + ci * cA);

      float sB, cB; __sincosf(bk * pB, &sB, &cB);
      float eB = __expf(ak * pB);
      float B1 = eB * cB;
      float B2 = eB * sB;

      if (t == 0) { a1.x = A1; a2.x = A2; b1.x = B1; b2.x = B2; }
      else        { a1.y = A1; a2.y = A2; b1.y = B1; b2.y = B2; }
    }
    // acc += A1*B1 ; acc += A2*B2   (f32 matrix pipe, full precision)
    acc = __builtin_amdgcn_wmma_f32_16x16x4_f32(false, a1, false, b1,
                                                (short)0, acc, false, false);
    acc = __builtin_amdgcn_wmma_f32_16x16x4_f32(false, a2, false, b2,
                                                (short)0, acc, false, false);
  }

  // D layout: VGPR r, lanes 0-15 -> M=r, lanes 16-31 -> M=r+8; N = lane&15.
#pragma unroll
  for (int r = 0; r < 8; ++r) {
    int M = r + 8 * half;
    out[l0 + M * 16 + idx16] = acc[r];
  }
}

// ---------------------------------------------------------------------------
extern "C" void kernel_launch(void* const* d_in, const int* in_sizes, int n_in,
                              void* d_out, int out_size, void* d_ws, size_t ws_size,
                              hipStream_t stream) {
  const float* W        = (const float*)d_in[0];  // (1, 256, 2)
  const float* log_step = (const float*)d_in[1];  // (1,)
  const float* Lre      = (const float*)d_in[2];  // (256,)
  const float* Lim      = (const float*)d_in[3];  // (256,)
  float* ws  = (float*)d_ws;                      // 4 KB used
  float* out = (float*)d_out;                     // (65536,)

  dss_prep_kernel<<<1, N_CH, 0, stream>>>(W, log_step, Lre, Lim, ws);
  // 256 tiles of 256 outputs; 8 waves (tiles) per 256-thread block -> 32 blocks.
  dss_wmma_kernel<<<L_TOT / 2048, 256, 0, stream>>>(ws, out);
}